// HardSamplingLayer_5360119186055
// MI455X (gfx1250) — compile-verified
//
#include <hip/hip_runtime.h>
#include <cstdint>

// HardSamplingLayer gather: out[t*256 + j] = x[t*1024 + w[j]], t = b*64 + p.
// Pure memory op (0 FLOPs). Strategy: stream each 4KB x-tile into LDS with
// CDNA5 async global->LDS loads (ASYNCcnt-tracked), gather randomly from LDS
// (ds_load_b32), emit coalesced b128 stores. Double-buffered per wave32 so no
// workgroup barriers are needed.

constexpr int L   = 1024;                 // floats per tile (dense_sampling_L)
constexpr int LT  = 256;                  // gathered cols per tile
constexpr int NP  = 64;                   // num_of_adc_p
constexpr int NB  = 4096;                 // BATCH
constexpr int TILES = NB * NP;            // 262144
constexpr int WAVES_PER_BLOCK = 8;
constexpr int BLOCK = 256;
constexpr int GRID  = 8192;               // 65536 waves -> 4 tiles per wave

// Issue one 4KB tile (1024 floats) as 8 async b128 loads. GVS addressing:
// mem = SGPR64(xbase) + VGPR32(vofs) + inst_offset ; lds = VGPR32(ldsAddr) + inst_offset.
__device__ __forceinline__ void issue_tile(uint32_t ldsAddr, const float* xbase,
                                           uint32_t vofs) {
  asm volatile(
      "global_load_async_to_lds_b128 %0, %1, %2 offset:0\n\t"
      "global_load_async_to_lds_b128 %0, %1, %2 offset:512\n\t"
      "global_load_async_to_lds_b128 %0, %1, %2 offset:1024\n\t"
      "global_load_async_to_lds_b128 %0, %1, %2 offset:1536\n\t"
      "global_load_async_to_lds_b128 %0, %1, %2 offset:2048\n\t"
      "global_load_async_to_lds_b128 %0, %1, %2 offset:2560\n\t"
      "global_load_async_to_lds_b128 %0, %1, %2 offset:3072\n\t"
      "global_load_async_to_lds_b128 %0, %1, %2 offset:3584"
      :
      : "v"(ldsAddr), "v"(vofs), "s"(xbase)
      : "memory");
}

__global__ __launch_bounds__(BLOCK) void hard_sampling_gather(
    const float* __restrict__ x, const long long* __restrict__ w64,
    float* __restrict__ out) {
  // Per-wave double buffer: 8 waves * 2 * 4KB = 64KB static LDS.
  __shared__ float smem[WAVES_PER_BLOCK * 2 * L];

  const int lane = threadIdx.x & 31;
  const int warp = threadIdx.x >> 5;

  // Preload sampling offsets (int64 -> u32 element index), 8 per lane.
  // Lane l owns output columns j = l*8 .. l*8+7 (contiguous 32B per lane).
  uint32_t widx[8];
#pragma unroll
  for (int k = 0; k < 8; ++k) widx[k] = (uint32_t)w64[lane * 8 + k];

  float* buf0 = &smem[warp * 2 * L];
  float* buf1 = buf0 + L;
  const uint32_t ldsA = (uint32_t)(uintptr_t)buf0 + (uint32_t)lane * 16u;
  const uint32_t ldsB = (uint32_t)(uintptr_t)buf1 + (uint32_t)lane * 16u;

  const int wid    = blockIdx.x * WAVES_PER_BLOCK + warp;
  const int stride = gridDim.x * WAVES_PER_BLOCK;

  int tile = wid;
  int cur  = 0;
  if (tile < TILES)
    issue_tile(ldsA, x, (uint32_t)tile * (uint32_t)(L * 4) + (uint32_t)lane * 16u);

  for (; tile < TILES; tile += stride) {
    const int next = tile + stride;
    if (next < TILES) {
      // Prefetch next tile into the other buffer, then wait for current
      // tile's 8 loads (async loads complete in order -> cnt<=8 suffices).
      issue_tile(cur ? ldsA : ldsB, x,
                 (uint32_t)next * (uint32_t)(L * 4) + (uint32_t)lane * 16u);
      asm volatile("s_wait_asynccnt 8" ::: "memory");
    } else {
      asm volatile("s_wait_asynccnt 0" ::: "memory");
    }

    const float* cb = cur ? buf1 : buf0;   // LDS pointer -> ds_load_b32 gathers
    float v[8];
#pragma unroll
    for (int k = 0; k < 8; ++k) v[k] = cb[widx[k]];

    // Output tile is 256 contiguous floats; lane l writes 32B at base+32*l
    // as two b128 stores -> fully coalesced 1KB.
    float4* o4 = (float4*)(out + (size_t)tile * LT + (size_t)lane * 8);
    o4[0] = make_float4(v[0], v[1], v[2], v[3]);
    o4[1] = make_float4(v[4], v[5], v[6], v[7]);

    cur ^= 1;
  }
}

extern "C" void kernel_launch(void* const* d_in, const int* in_sizes, int n_in,
                              void* d_out, int out_size, void* d_ws, size_t ws_size,
                              hipStream_t stream) {
  const float* x       = (const float*)d_in[0];       // [4096, 65536] f32
  const long long* w   = (const long long*)d_in[1];   // [256] i64
  float* out           = (float*)d_out;               // [4096, 16384] f32
  (void)in_sizes; (void)n_in; (void)out_size; (void)d_ws; (void)ws_size;

  hipLaunchKernelGGL(hard_sampling_gather, dim3(GRID), dim3(BLOCK), 0, stream,
                     x, w, out);
}